// _CliffordBatchNorm_31413390803045
// MI455X (gfx1250) — compile-verified
//
#include <hip/hip_runtime.h>

typedef __attribute__((ext_vector_type(2))) float v2f;
typedef __attribute__((ext_vector_type(8))) float v8f;

// Problem constants (match reference setup_inputs)
#define BB 32
#define CC 64
#define HH 64
#define WW 64
#define II 4
#define POS_PER_TILE 4096              // H*W positions per (b,c) tile
#define FLOATS_PER_TILE (POS_PER_TILE*II)  // 16384 floats = 64KB
#define NBC (BB*CC)                    // 2048 tiles
#define NTOT (BB*HH*WW)                // 131072 samples per channel
#define EPS 1e-5f

// ---------------------------------------------------------------------------
// Kernel 1: per-(b,c) partial sums: s[0..3] = sum x_i, q[0..9] = sum x_i x_j
// ---------------------------------------------------------------------------
__global__ void __launch_bounds__(256)
cbn_stats(const float* __restrict__ x, float* __restrict__ part) {
    const int bc = blockIdx.x;
    const size_t base = (size_t)bc * FLOATS_PER_TILE;
    const int t = threadIdx.x;

    float a0=0.f,a1=0.f,a2=0.f,a3=0.f;
    float q00=0.f,q01=0.f,q02=0.f,q03=0.f,q11=0.f,q12=0.f,q13=0.f,q22=0.f,q23=0.f,q33=0.f;

    for (int p = t; p < POS_PER_TILE; p += 256) {
        const float4 v = *(const float4*)(x + base + (size_t)p * II);
        a0 += v.x; a1 += v.y; a2 += v.z; a3 += v.w;
        q00 = fmaf(v.x, v.x, q00); q01 = fmaf(v.x, v.y, q01);
        q02 = fmaf(v.x, v.z, q02); q03 = fmaf(v.x, v.w, q03);
        q11 = fmaf(v.y, v.y, q11); q12 = fmaf(v.y, v.z, q12);
        q13 = fmaf(v.y, v.w, q13); q22 = fmaf(v.z, v.z, q22);
        q23 = fmaf(v.z, v.w, q23); q33 = fmaf(v.w, v.w, q33);
    }

    float acc[14] = {a0,a1,a2,a3,q00,q01,q02,q03,q11,q12,q13,q22,q23,q33};

    __shared__ float red[14 * 256];
    #pragma unroll
    for (int k = 0; k < 14; ++k) red[k * 256 + t] = acc[k];
    __syncthreads();
    for (int s = 128; s > 0; s >>= 1) {
        if (t < s) {
            #pragma unroll
            for (int k = 0; k < 14; ++k)
                red[k * 256 + t] += red[k * 256 + t + s];
        }
        __syncthreads();
    }
    if (t == 0) {
        #pragma unroll
        for (int k = 0; k < 14; ++k) part[(size_t)bc * 16 + k] = red[k * 256];
    }
}

// ---------------------------------------------------------------------------
// Kernel 2: per-channel finalize: reduce partials -> cov -> chol -> M, b2
//   M_c = W_c * U^-1  (solved row-wise against U = L^T),  b2 = bias - M*mean
// ---------------------------------------------------------------------------
__global__ void __launch_bounds__(64)
cbn_finalize(const float* __restrict__ part, const float* __restrict__ weight,
             const float* __restrict__ bias, float* __restrict__ Mbuf,
             float* __restrict__ Bbuf) {
    const int c = threadIdx.x;
    if (c >= CC) return;

    float acc[14];
    #pragma unroll
    for (int k = 0; k < 14; ++k) acc[k] = 0.f;
    for (int b = 0; b < BB; ++b) {               // fixed order -> deterministic
        const float* p = part + (size_t)(b * CC + c) * 16;
        #pragma unroll
        for (int k = 0; k < 14; ++k) acc[k] += p[k];
    }

    const float invN = 1.0f / (float)NTOT;
    float mean[4];
    #pragma unroll
    for (int i = 0; i < 4; ++i) mean[i] = acc[i] * invN;

    float cov[4][4];
    {
        int idx = 4;
        for (int i = 0; i < 4; ++i)
            for (int j = i; j < 4; ++j) {
                float v = acc[idx++] * invN - mean[i] * mean[j];
                cov[i][j] = v; cov[j][i] = v;
            }
        for (int i = 0; i < 4; ++i) cov[i][i] += EPS;
    }

    // Cholesky: cov = L * L^T (lower)
    float L[4][4];
    for (int i = 0; i < 4; ++i)
        for (int j = 0; j < 4; ++j) L[i][j] = 0.f;
    for (int i = 0; i < 4; ++i) {
        for (int j = 0; j <= i; ++j) {
            float s = cov[i][j];
            for (int k = 0; k < j; ++k) s -= L[i][k] * L[j][k];
            L[i][j] = (i == j) ? sqrtf(s) : s / L[j][j];
        }
    }

    // Solve m * U = w per row of W (U[k][j] = L[j][k]); forward substitution.
    float M[4][4];
    for (int i = 0; i < 4; ++i) {
        float m[4];
        for (int j = 0; j < 4; ++j) {
            float s = weight[(size_t)(i * 4 + j) * CC + c];
            for (int k = 0; k < j; ++k) s -= m[k] * L[j][k];
            m[j] = s / L[j][j];
        }
        for (int j = 0; j < 4; ++j) M[i][j] = m[j];
    }

    float b2[4];
    for (int i = 0; i < 4; ++i) {
        float s = bias[(size_t)i * CC + c];
        for (int j = 0; j < 4; ++j) s -= M[i][j] * mean[j];
        b2[i] = s;
    }

    for (int i = 0; i < 4; ++i)
        for (int j = 0; j < 4; ++j) Mbuf[(size_t)c * 16 + i * 4 + j] = M[i][j];
    for (int i = 0; i < 4; ++i) Bbuf[(size_t)c * 4 + i] = b2[i];
}

// ---------------------------------------------------------------------------
// Kernel 3: apply  out = M_c * x + b2_c  via V_WMMA_F32_16X16X4_F32
//   A(16x4)  = 16 positions' 4-vectors (rows = positions)
//   B(4x16)  = M^T in columns 0..3
//   C(16x16) = broadcast b2 per column  (bias folded into accumulator)
//   D cols 0..3 -> repacked through LDS -> coalesced 256B NT stores per wave
// ---------------------------------------------------------------------------
__global__ void __launch_bounds__(256)
cbn_apply(const float* __restrict__ x, const float* __restrict__ Mbuf,
          const float* __restrict__ Bbuf, float* __restrict__ out) {
    const int bc   = blockIdx.x;           // 0..2047  (b*C + c)
    const int c    = bc & (CC - 1);
    const size_t base = (size_t)bc * FLOATS_PER_TILE;
    const int lane = threadIdx.x & 31;
    const int wave = threadIdx.x >> 5;     // 0..7
    const int n    = lane & 15;
    const bool hi  = lane >= 16;

    // B operand: lanes 0..15 hold rows K=0 (lo half) / K=2 (hi half);
    // second VGPR holds K=1 / K=3.  B[k][col] = M^T[k][col] = M[col][k].
    v2f breg; breg.x = 0.f; breg.y = 0.f;
    if (n < 4) {
        const int k0 = hi ? 2 : 0;
        breg.x = Mbuf[(size_t)c * 16 + n * 4 + k0];
        breg.y = Mbuf[(size_t)c * 16 + n * 4 + k0 + 1];
    }
    // C init: column i of D gets b2[i] in every row.
    const float bv = (n < 4) ? Bbuf[(size_t)c * 4 + n] : 0.f;
    v8f cinit = {bv, bv, bv, bv, bv, bv, bv, bv};

    __shared__ float lds[8 * 64];          // 64 floats (16 pos x 4 comps) per wave
    float* wlds = &lds[wave * 64];

    #pragma unroll 2
    for (int iter = 0; iter < 32; ++iter) {
        const int tile = iter * 8 + wave;                 // 16-position tile id
        const size_t toff = base + (size_t)tile * 64;     // floats

        // A operand: row p = lane%16; lo half = comps {0,1}, hi half = {2,3}.
        // Whole wave covers a contiguous 256B block.
        const v2f a = *(const v2f*)(x + toff + (size_t)n * 4 + (hi ? 2 : 0));

        v8f d = __builtin_amdgcn_wmma_f32_16x16x4_f32(
            false, a, false, breg, (short)0, cinit, false, false);

        // Repack D cols 0..3 into LDS in memory order [p*4 + i]
        if (n < 4) {
            const int prow = hi ? 8 : 0;
            #pragma unroll
            for (int v = 0; v < 8; ++v)
                wlds[(prow + v) * 4 + n] = d[v];
        }
        asm volatile("s_wait_dscnt 0" ::: "memory");   // DS in-order per wave; explicit fence

        // Coalesced readback + 256B contiguous NT store per wave
        const v2f o = *(const v2f*)&wlds[lane * 2];
        __builtin_nontemporal_store(o, (v2f*)(out + toff + (size_t)lane * 2));
    }
}

// ---------------------------------------------------------------------------
extern "C" void kernel_launch(void* const* d_in, const int* in_sizes, int n_in,
                              void* d_out, int out_size, void* d_ws, size_t ws_size,
                              hipStream_t stream) {
    const float* x      = (const float*)d_in[0];   // (32,64,64,64,4) f32
    const float* weight = (const float*)d_in[1];   // (4,4,64) f32
    const float* bias   = (const float*)d_in[2];   // (4,64) f32
    float* out = (float*)d_out;

    // Workspace layout (floats): partials[2048*16] | M[64*16] | b2[64*4]
    float* part = (float*)d_ws;
    float* Mbuf = part + (size_t)NBC * 16;
    float* Bbuf = Mbuf + (size_t)CC * 16;

    cbn_stats   <<<NBC, 256, 0, stream>>>(x, part);
    cbn_finalize<<<1,    64, 0, stream>>>(part, weight, bias, Mbuf, Bbuf);
    cbn_apply   <<<NBC, 256, 0, stream>>>(x, Mbuf, Bbuf, out);
}